// Encoder_64244120814048
// MI455X (gfx1250) — compile-verified
//
#include <hip/hip_runtime.h>

#define N_NODES 100000
#define N_EDGES 1600000
#define N_FEAT  256
#define HIDDEN  128

typedef __attribute__((ext_vector_type(2))) float v2f;
typedef __attribute__((ext_vector_type(8))) float v8f;

// ---------- degree / normalization ----------
__global__ void deg_init_kernel(float* __restrict__ deg) {
    int i = blockIdx.x * blockDim.x + threadIdx.x;
    if (i < N_NODES) deg[i] = 1.0f;   // self-loop contributes 1 to every degree
}

__global__ void deg_count_kernel(const long long* __restrict__ ei, float* __restrict__ deg) {
    int e = blockIdx.x * blockDim.x + threadIdx.x;
    if (e < N_EDGES) {
        long long dst = ei[N_EDGES + e];   // col = target index
        __hip_atomic_fetch_add(&deg[dst], 1.0f, __ATOMIC_RELAXED, __HIP_MEMORY_SCOPE_AGENT);
    }
}

__global__ void dinv_kernel(float* __restrict__ deg) {
    int i = blockIdx.x * blockDim.x + threadIdx.x;
    if (i < N_NODES) deg[i] = rsqrtf(deg[i]);   // deg >= 1 always
}

// ---------- h = x @ W^T via fp32 WMMA (16x16x4) ----------
__global__ void __launch_bounds__(256) gemm_kernel(const float* __restrict__ x,
                                                   const float* __restrict__ W,
                                                   float* __restrict__ h) {
    __shared__ float xs[16 * N_FEAT];   // 16 KB tile of x
    const int t     = threadIdx.x;
    const int node0 = blockIdx.x * 16;

    // coalesced flat copy of 16 consecutive rows (4096 floats = 1024 float4)
    const float4* xg = (const float4*)(x + (size_t)node0 * N_FEAT);
    float4* xl = (float4*)xs;
#pragma unroll
    for (int i = 0; i < 4; ++i) xl[t + 256 * i] = xg[t + 256 * i];
    __syncthreads();

    const int wave  = t >> 5;
    const int lane  = t & 31;
    const int mn    = lane & 15;            // A row (m) == B col (n) for this lane
    const int khalf = (lane >> 4) << 1;     // lanes 16-31 hold K = k+2, k+3
    const int o0    = wave * 16;            // 8 waves cover HIDDEN = 128

    const float* arow = xs + mn * N_FEAT;
    const float* wrow = W + (size_t)(o0 + mn) * N_FEAT;

    v8f c = {};
#pragma unroll 4
    for (int k = 0; k < N_FEAT; k += 4) {
        v2f a = *(const v2f*)(arow + k + khalf);   // A: 16x4 fp32 fragment
        v2f b = *(const v2f*)(wrow + k + khalf);   // B: 4x16 fp32 fragment (W^T)
        c = __builtin_amdgcn_wmma_f32_16x16x4_f32(false, a, false, b,
                                                  (short)0, c, false, false);
    }

    // C/D layout: lanes 0-15 -> M = r, lanes 16-31 -> M = 8 + r; N = lane & 15
    const int mbase = (lane >> 4) * 8;
#pragma unroll
    for (int r = 0; r < 8; ++r)
        h[(size_t)(node0 + mbase + r) * HIDDEN + o0 + mn] = c[r];
}

// ---------- out = h * dinv^2 (self-loop term; also initializes d_out) ----------
__global__ void selfloop_kernel(const float* __restrict__ h,
                                const float* __restrict__ dinv,
                                float* __restrict__ out) {
    int tid = blockIdx.x * blockDim.x + threadIdx.x;   // < 12.8M
    if (tid < N_NODES * HIDDEN) {
        int i = tid >> 7;                 // node index (HIDDEN = 128)
        float d = dinv[i];
        out[tid] = h[tid] * d * d;
    }
}

// ---------- edge scatter: one wave per edge ----------
__global__ void __launch_bounds__(256) scatter_kernel(const long long* __restrict__ ei,
                                                      const float* __restrict__ h,
                                                      const float* __restrict__ dinv,
                                                      float* __restrict__ agg) {
    int wid  = (blockIdx.x * blockDim.x + threadIdx.x) >> 5;   // edge id
    int lane = threadIdx.x & 31;
    if (wid >= N_EDGES) return;

    long long src = ei[wid];
    long long dst = ei[N_EDGES + wid];
    float nrm = dinv[src] * dinv[dst];

    float4 v = ((const float4*)(h + (size_t)src * HIDDEN))[lane];  // coalesced 512B gather
    float* o = agg + (size_t)dst * HIDDEN + lane * 4;
    __hip_atomic_fetch_add(o + 0, v.x * nrm, __ATOMIC_RELAXED, __HIP_MEMORY_SCOPE_AGENT);
    __hip_atomic_fetch_add(o + 1, v.y * nrm, __ATOMIC_RELAXED, __HIP_MEMORY_SCOPE_AGENT);
    __hip_atomic_fetch_add(o + 2, v.z * nrm, __ATOMIC_RELAXED, __HIP_MEMORY_SCOPE_AGENT);
    __hip_atomic_fetch_add(o + 3, v.w * nrm, __ATOMIC_RELAXED, __HIP_MEMORY_SCOPE_AGENT);
}

// ---------- bias + PReLU (in place on d_out) ----------
__global__ void finalize_kernel(const float* __restrict__ b,
                                const float* __restrict__ pa,
                                float* __restrict__ out) {
    int tid = blockIdx.x * blockDim.x + threadIdx.x;
    if (tid < N_NODES * HIDDEN) {
        int c = tid & (HIDDEN - 1);
        float v = out[tid] + b[c];
        out[tid] = v > 0.0f ? v : pa[c] * v;
    }
}

extern "C" void kernel_launch(void* const* d_in, const int* in_sizes, int n_in,
                              void* d_out, int out_size, void* d_ws, size_t ws_size,
                              hipStream_t stream) {
    const float*     x  = (const float*)d_in[0];
    const long long* ei = (const long long*)d_in[1];   // int64 edge_index [2, E]
    const float*     W  = (const float*)d_in[2];
    const float*     b  = (const float*)d_in[3];
    const float*     pa = (const float*)d_in[4];
    float* out = (float*)d_out;

    float* h   = (float*)d_ws;                                        // 51.2 MB
    float* deg = (float*)((char*)d_ws + (size_t)N_NODES * HIDDEN * sizeof(float)); // 400 KB (deg -> dinv)

    deg_init_kernel <<<(N_NODES + 255) / 256, 256, 0, stream>>>(deg);
    deg_count_kernel<<<(N_EDGES + 255) / 256, 256, 0, stream>>>(ei, deg);
    dinv_kernel     <<<(N_NODES + 255) / 256, 256, 0, stream>>>(deg);

    gemm_kernel     <<<N_NODES / 16, 256, 0, stream>>>(x, W, h);      // 100000 % 16 == 0

    selfloop_kernel <<<(N_NODES * HIDDEN + 255) / 256, 256, 0, stream>>>(h, deg, out);
    scatter_kernel  <<<(N_EDGES * 32 + 255) / 256, 256, 0, stream>>>(ei, h, deg, out);
    finalize_kernel <<<(N_NODES * HIDDEN + 255) / 256, 256, 0, stream>>>(b, pa, out);
}